// Loss_26139170963548
// MI455X (gfx1250) — compile-verified
//
#include <hip/hip_runtime.h>
#include <hip/hip_bf16.h>

#define N_  512
#define D_  256
#define DY_ 256
#define K_  16
#define NK  (N_ * K_)     // 8192 gathered rows
#define H_  512

#define EPSF 1e-12f
#define WF   0.7f
#define W1F  0.3f         // 1 - WEIGHT

typedef __attribute__((ext_vector_type(2))) float v2f;
typedef __attribute__((ext_vector_type(8))) float v8f;
typedef __attribute__((ext_vector_type(4))) unsigned int v4u;
typedef __attribute__((ext_vector_type(8))) int v8i;
typedef __attribute__((ext_vector_type(4))) int v4i;

#if defined(__has_builtin)
#if __has_builtin(__builtin_amdgcn_tensor_load_to_lds) && __has_builtin(__builtin_amdgcn_s_wait_tensorcnt)
#define HAVE_TDM 1
#endif
#endif
#ifndef HAVE_TDM
#define HAVE_TDM 0
#endif

// ---- workspace layout (float offsets), ~20 MB total ----
enum : size_t {
  WS_CT    = 0,
  WS_NBSUM = 1,
  WS_XN2   = 64,
  WS_YN2   = WS_XN2 + N_,
  WS_NF2   = WS_YN2 + N_,
  WS_ON2   = WS_NF2 + NK,
  WS_DX    = WS_ON2 + NK,
  WS_DY    = WS_DX + NK,
  WS_SX    = WS_DY + NK,
  WS_SY    = WS_SX + (size_t)N_ * N_,
  WS_NB    = WS_SY + (size_t)N_ * N_,
  WS_NF    = WS_NB + (size_t)N_ * N_,        // 8192 x 256
  WS_ON    = WS_NF + (size_t)NK * D_,        // 8192 x 256
};

// ---- f32 WMMA 16x16x4: D = A(16x4) * B(4x16) + C ----
__device__ __forceinline__ v8f wmma4(v2f a, v2f b, v8f c) {
  return __builtin_amdgcn_wmma_f32_16x16x4_f32(
      /*neg_a=*/false, a, /*neg_b=*/false, b,
      /*c_mod=*/(short)0, c, /*reuse_a=*/false, /*reuse_b=*/false);
}

// A operand: A[m,kk] = src[row0+m][k0+kk], src row-major, leading dim ld.
// lanes 0-15: m=lane, kk=0,1 ; lanes 16-31: m=lane-16, kk=2,3  -> contiguous float2
__device__ __forceinline__ v2f load_a(const float* src, int ld, int row0, int k0, int lane) {
  int m  = lane & 15;
  int kk = (lane >> 4) << 1;
  const float* q = src + (size_t)(row0 + m) * ld + (k0 + kk);
  v2f r; r.x = q[0]; r.y = q[1]; return r;
}

// B operand for B = tile-of-src-transposed: B[kk,nn] = src[col0+nn][k0+kk].
__device__ __forceinline__ v2f load_bt(const float* src, int ld, int col0, int k0, int lane) {
  return load_a(src, ld, col0, k0, lane);
}

// B operand for row-major weight W[K,Nd]: B[kk,nn] = W[k0+kk][n0+nn]
__device__ __forceinline__ v2f load_b_rm(const float* W, int ld, int k0, int n0, int lane) {
  int nn = lane & 15;
  int kk = (lane >> 4) << 1;
  const float* q = W + (size_t)(k0 + kk) * ld + (n0 + nn);
  v2f r; r.x = q[0]; r.y = q[ld]; return r;
}

#if HAVE_TDM
// ---- Tensor Data Mover: contiguous 1-D tile (nelem f32) global -> LDS ----
__device__ __forceinline__ void tdm_load_linear(unsigned int lds_addr, const void* gptr,
                                                unsigned int nelem) {
  unsigned long long ga = (unsigned long long)(uintptr_t)gptr;
  v4u g0;
  g0.x = 1u;                                                // count=1, user mode
  g0.y = lds_addr;                                          // lds_addr [63:32]
  g0.z = (unsigned int)ga;                                  // global_addr lo
  g0.w = (unsigned int)((ga >> 32) & 0x1FFFFFFu) | (2u << 30);  // addr hi | type=2
  v8i g1;
  g1[0] = (int)(2u << 16);                                  // data_size = 4B
  g1[1] = (int)((nelem & 0xFFFFu) << 16);                   // tensor_dim0[15:0]
  g1[2] = (int)(((nelem >> 16) & 0xFFFFu) | (1u << 16));    // tensor_dim0[31:16] | tensor_dim1=1
  g1[3] = (int)((nelem & 0xFFFFu) << 16);                   // tile_dim0 = nelem
  g1[4] = 0;                                                // tile_dim1/2 unused
  g1[5] = (int)nelem;                                       // tensor_dim0_stride lo32
  g1[6] = 0;
  g1[7] = 0;
  v4i z4 = {0, 0, 0, 0};
  v8i z8 = {0, 0, 0, 0, 0, 0, 0, 0};
  __builtin_amdgcn_tensor_load_to_lds(g0, g1, z4, z4, z8, 0);
}

// ---- TDM gather mode: 16 rows of `rowlen` f32, 16-bit row indices ----
__device__ __forceinline__ void tdm_gather16(unsigned int lds_addr, const void* base,
                                             unsigned int rowlen, unsigned int nrows_tensor,
                                             const int* idx) {
  unsigned long long ga = (unsigned long long)(uintptr_t)base;
  v4u g0;
  g0.x = 1u | (0u << 30) | (1u << 31);                      // count=1, u16 idx, gather_mode
  g0.y = lds_addr;
  g0.z = (unsigned int)ga;
  g0.w = (unsigned int)((ga >> 32) & 0x1FFFFFFu) | (2u << 30);
  v8i g1;
  g1[0] = (int)(2u << 16);                                  // data_size = 4B
  g1[1] = (int)((rowlen & 0xFFFFu) << 16);                  // tensor_dim0[15:0]
  g1[2] = (int)(((rowlen >> 16) & 0xFFFFu) | ((nrows_tensor & 0xFFFFu) << 16));
  g1[3] = (int)(((nrows_tensor >> 16) & 0xFFFFu) | ((rowlen & 0xFFFFu) << 16)); // tile_dim0
  g1[4] = 16;                                               // tile_dim1 = #valid indices
  g1[5] = (int)rowlen;                                      // tensor_dim0_stride lo32
  g1[6] = 0;
  g1[7] = 0;
  v4i g2, g3;
#pragma unroll
  for (int w = 0; w < 4; ++w) {
    g2[w] = (idx[2 * w] & 0xFFFF) | ((idx[2 * w + 1] & 0xFFFF) << 16);
    g3[w] = (idx[8 + 2 * w] & 0xFFFF) | ((idx[8 + 2 * w + 1] & 0xFFFF) << 16);
  }
  v8i z8 = {0, 0, 0, 0, 0, 0, 0, 0};
  __builtin_amdgcn_tensor_load_to_lds(g0, g1, g2, g3, z8, 0);
}
#endif  // HAVE_TDM

// ---------------- kernel 1: zero accumulators + row norms ----------------
__global__ void k_init(const float* __restrict__ x, const float* __restrict__ y,
                       float* __restrict__ ws) {
  int t = threadIdx.x;  // 512 threads
  if (t == 0) { ws[WS_CT] = 0.f; ws[WS_NBSUM] = 0.f; }
  float sx = 0.f, sy = 0.f;
  const float* xr = x + (size_t)t * D_;
  const float* yr = y + (size_t)t * DY_;
  for (int d = 0; d < D_; ++d)  { float a = xr[d]; sx += a * a; }
  for (int d = 0; d < DY_; ++d) { float b = yr[d]; sy += b * b; }
  ws[WS_XN2 + t] = sx;
  ws[WS_YN2 + t] = sy;
}

// ---------------- kernel 2: gather + 2-layer MLP on 8192 rows ----------------
// block j (512 blocks, 128 threads): rows c = j*16 .. j*16+15
__global__ void k_mlp(const float* __restrict__ dataset, const int* __restrict__ graph,
                      const float* __restrict__ W1, const float* __restrict__ W2,
                      float* __restrict__ ws) {
  __shared__ __align__(16) float s_nf[16 * D_];   // 16 KB
  __shared__ __align__(16) float s_h[16 * H_];    // 32 KB
  int j    = blockIdx.x;
  int tid  = threadIdx.x;
  int lane = tid & 31, wv = tid >> 5;
  float* nf_ws = ws + WS_NF + (size_t)j * K_ * D_;

#if HAVE_TDM
  // one TDM gather-mode DMA fetches all 16 neighbor rows into LDS
  if (wv == 0) {
    int idx[16];
#pragma unroll
    for (int r = 0; r < 16; ++r) idx[r] = graph[j * K_ + r];
    tdm_gather16((unsigned int)(uintptr_t)s_nf, dataset, D_, 20000u, idx);
    __builtin_amdgcn_s_wait_tensorcnt((short)0);
  }
  __syncthreads();
  // mirror gathered tile to workspace (consumed by k_ct as B operand)
  for (int v4 = tid; v4 < 16 * D_ / 4; v4 += 128)
    ((float4*)nf_ws)[v4] = ((const float4*)s_nf)[v4];
#else
  for (int v4 = tid; v4 < 16 * D_ / 4; v4 += 128) {
    int r  = v4 >> 6;            // 64 float4 per row
    int c4 = v4 & 63;
    int g  = graph[j * K_ + r];
    float4 val = ((const float4*)(dataset + (size_t)g * D_))[c4];
    ((float4*)s_nf)[v4]  = val;
    ((float4*)nf_ws)[v4] = val;
  }
  __syncthreads();
#endif

  // layer 1: hidden[16,512] = relu(nf @ W1), 8 n-tiles per wave
  for (int nt = wv; nt < H_ / 16; nt += 4) {
    v8f acc = {};
#pragma unroll 4
    for (int k0 = 0; k0 < D_; k0 += 4)
      acc = wmma4(load_a(s_nf, D_, 0, k0, lane),
                  load_b_rm(W1, H_, k0, nt * 16, lane), acc);
    int ncol = lane & 15, rb = (lane >> 4) * 8;
#pragma unroll
    for (int r = 0; r < 8; ++r) {
      float v = acc[r];
      s_h[(size_t)(rb + r) * H_ + nt * 16 + ncol] = v > 0.f ? v : 0.f;
    }
  }
  __syncthreads();

  // layer 2: on[16,256] = hidden @ W2, 4 n-tiles per wave
  float* on_ws = ws + WS_ON + (size_t)j * K_ * DY_;
  for (int nt = wv; nt < DY_ / 16; nt += 4) {
    v8f acc = {};
#pragma unroll 4
    for (int k0 = 0; k0 < H_; k0 += 4)
      acc = wmma4(load_a(s_h, H_, 0, k0, lane),
                  load_b_rm(W2, DY_, k0, nt * 16, lane), acc);
    int ncol = lane & 15, rb = (lane >> 4) * 8;
#pragma unroll
    for (int r = 0; r < 8; ++r)
      on_ws[(size_t)(rb + r) * DY_ + nt * 16 + ncol] = acc[r];
  }
}

// ---------------- kernel 3: per-row stats ----------------
__global__ void k_stats(const float* __restrict__ x, const float* __restrict__ y,
                        float* __restrict__ ws) {
  int c = blockIdx.x * blockDim.x + threadIdx.x;
  if (c >= NK) return;
  int j = c >> 4;
  const float* nf = ws + WS_NF + (size_t)c * D_;
  const float* on = ws + WS_ON + (size_t)c * DY_;
  const float* xj = x + (size_t)j * D_;
  const float* yj = y + (size_t)j * DY_;
  float s_nf2 = 0.f, s_on2 = 0.f, s_dx = 0.f, s_dy = 0.f;
  for (int d = 0; d < D_; ++d)  { float a = nf[d]; s_nf2 += a * a; s_dx += xj[d] * a; }
  for (int d = 0; d < DY_; ++d) { float b = on[d]; s_on2 += b * b; s_dy += yj[d] * b; }
  ws[WS_NF2 + c] = s_nf2;  ws[WS_ON2 + c] = s_on2;
  ws[WS_DX + c]  = s_dx;   ws[WS_DY + c]  = s_dy;
}

// ---------------- kernel 4: Gram matrices + nb_loss epilogue ----------------
// 256 blocks x 128 threads; each wave computes one 16x16 tile of Sx AND Sy
__global__ void k_gram(const float* __restrict__ x, const float* __restrict__ y,
                       float* __restrict__ ws) {
  __shared__ float blocksum;
  int tid = threadIdx.x, lane = tid & 31, wv = tid >> 5;
  if (tid == 0) blocksum = 0.f;
  __syncthreads();

  int t  = blockIdx.x * 4 + wv;    // 0..1023
  int i0 = (t >> 5) * 16, j0 = (t & 31) * 16;

  v8f ax = {}, ay = {};
#pragma unroll 4
  for (int k0 = 0; k0 < D_; k0 += 4) {
    ax = wmma4(load_a(x, D_, i0, k0, lane), load_bt(x, D_, j0, k0, lane), ax);
    ay = wmma4(load_a(y, DY_, i0, k0, lane), load_bt(y, DY_, j0, k0, lane), ay);
  }

  int ncol = lane & 15, rb = (lane >> 4) * 8;
  int jcol = j0 + ncol;
  float xnj = ws[WS_XN2 + jcol], ynj = ws[WS_YN2 + jcol];
  float part = 0.f;
#pragma unroll
  for (int r = 0; r < 8; ++r) {
    int irow = i0 + rb + r;
    float sx = ax[r], sy = ay[r];
    float xx = sqrtf(fmaxf(ws[WS_XN2 + irow] + xnj - 2.f * sx, EPSF));
    float yy = sqrtf(fmaxf(ws[WS_YN2 + irow] + ynj - 2.f * sy, EPSF));
    float nb = fmaxf(fabsf(xx - yy), 1e-6f);
    size_t o = (size_t)irow * N_ + jcol;
    ws[WS_SX + o] = sx;  ws[WS_SY + o] = sy;  ws[WS_NB + o] = nb;
    part += nb;
  }
  atomicAdd(&blocksum, part);
  __syncthreads();
  if (tid == 0) atomicAdd(&ws[WS_NBSUM], blocksum);
}

// ---------------- kernel 5: fused Gx/Gy GEMM + ct epilogue ----------------
// 1024 blocks (32 i-tiles x 32 j-tiles), 128 threads. Each wave handles 4 j's.
__global__ void k_ct(const float* __restrict__ x, const float* __restrict__ y,
                     float* __restrict__ ws) {
  __shared__ __align__(16) float s_x[16 * D_];    // 16 KB
  __shared__ __align__(16) float s_y[16 * DY_];   // 16 KB
  __shared__ float blocksum;
  int tid = threadIdx.x, lane = tid & 31, wv = tid >> 5;
  int it = blockIdx.x >> 5, jt = blockIdx.x & 31;
  int i0 = it * 16;
  if (tid == 0) blocksum = 0.f;

#if HAVE_TDM
  // stage the contiguous 16x256 x/y tiles with two TDM DMAs (wave 0 issues)
  if (wv == 0) {
    tdm_load_linear((unsigned int)(uintptr_t)s_x, x + (size_t)i0 * D_, 16u * D_);
    tdm_load_linear((unsigned int)(uintptr_t)s_y, y + (size_t)i0 * DY_, 16u * DY_);
    __builtin_amdgcn_s_wait_tensorcnt((short)0);
  }
  __syncthreads();
#else
  for (int v4 = tid; v4 < 16 * D_ / 4; v4 += 128) {
    int r = v4 >> 6, c4 = v4 & 63;
    ((float4*)s_x)[v4] = ((const float4*)(x + (size_t)(i0 + r) * D_))[c4];
    ((float4*)s_y)[v4] = ((const float4*)(y + (size_t)(i0 + r) * DY_))[c4];
  }
  __syncthreads();
#endif

  const float* nf = ws + WS_NF;
  const float* on = ws + WS_ON;
  float part = 0.f;

  for (int nt = wv; nt < 16; nt += 4) {
    int j  = jt * 16 + nt;          // this n-tile == all K neighbors of column j
    int c0 = j * K_;
    v8f gx = {}, gy = {};
#pragma unroll 4
    for (int k0 = 0; k0 < D_; k0 += 4) {
      gx = wmma4(load_a(s_x, D_, 0, k0, lane), load_bt(nf, D_, c0, k0, lane), gx);
      gy = wmma4(load_a(s_y, DY_, 0, k0, lane), load_bt(on, DY_, c0, k0, lane), gy);
    }
    int kcol = lane & 15, rb = (lane >> 4) * 8;
    int c = c0 + kcol;
    float nf2 = ws[WS_NF2 + c], on2 = ws[WS_ON2 + c];
    float dxd = ws[WS_DX + c],  dyd = ws[WS_DY + c];
    float xnj = ws[WS_XN2 + j], ynj = ws[WS_YN2 + j];
#pragma unroll
    for (int r = 0; r < 8; ++r) {
      int i = i0 + rb + r;
      if (i == j) continue;         // diagonal mask
      size_t o = (size_t)i * N_ + j;
      float cx2 = WF * WF * ws[WS_XN2 + i] + W1F * W1F * xnj + 2.f * WF * W1F * ws[WS_SX + o];
      float cy2 = WF * WF * ws[WS_YN2 + i] + W1F * W1F * ynj + 2.f * WF * W1F * ws[WS_SY + o];
      float dxv = sqrtf(fmaxf(cx2 + nf2 - 2.f * (WF * gx[r] + W1F * dxd), EPSF));
      float dyv = sqrtf(fmaxf(cy2 + on2 - 2.f * (WF * gy[r] + W1F * dyd), EPSF));
      part += fmaxf(fabsf(dxv - dyv), 1e-6f);
    }
  }
  atomicAdd(&blocksum, part);
  __syncthreads();
  if (tid == 0) atomicAdd(&ws[WS_CT], blocksum);
}

// ---------------- kernel 6: finalize outputs ----------------
__global__ void k_final(const float* __restrict__ ws, float* __restrict__ out) {
  float ct  = ws[WS_CT] / (float)(K_ * (size_t)N_ * N_);
  float nbm = ws[WS_NBSUM] / (float)((size_t)N_ * N_);
  size_t idx = (size_t)blockIdx.x * blockDim.x + threadIdx.x;
  if (idx == 0) { out[0] = nbm; out[1] = ct; }
  if (idx < (size_t)N_ * N_)
    out[2 + idx] = (ws[WS_NB + idx] + ct) * 0.5f;
}

extern "C" void kernel_launch(void* const* d_in, const int* in_sizes, int n_in,
                              void* d_out, int out_size, void* d_ws, size_t ws_size,
                              hipStream_t stream) {
  const float* x       = (const float*)d_in[0];
  const float* y       = (const float*)d_in[1];
  const float* dataset = (const float*)d_in[2];
  const float* W1      = (const float*)d_in[3];
  const float* W2      = (const float*)d_in[4];
  const int*   graph   = (const int*)d_in[5];
  float* ws  = (float*)d_ws;
  float* out = (float*)d_out;

  k_init <<<1, N_, 0, stream>>>(x, y, ws);
  k_mlp  <<<N_, 128, 0, stream>>>(dataset, graph, W1, W2, ws);
  k_stats<<<NK / 128, 128, 0, stream>>>(x, y, ws);
  k_gram <<<256, 128, 0, stream>>>(x, y, ws);
  k_ct   <<<1024, 128, 0, stream>>>(x, y, ws);
  k_final<<<((size_t)N_ * N_ + 255) / 256, 256, 0, stream>>>(ws, out);
}